// TensorProduct_19825569038933
// MI455X (gfx1250) — compile-verified
//
#include <hip/hip_runtime.h>

typedef float v2f __attribute__((ext_vector_type(2)));
typedef float v8f __attribute__((ext_vector_type(8)));

#define TP_BATCH 200000
#define TP_S1 384   // x1 row stride (floats)
#define TP_S2 256   // x2 row stride (floats)
#define TP_SO 257   // y  row stride (floats)

// ---------------------------------------------------------------------------
// Kernel 1: reduce weight[41088] into 5 per-path scalars (one block per path).
// ---------------------------------------------------------------------------
__global__ __launch_bounds__(256) void tp_weight_sums(const float* __restrict__ w,
                                                      float* __restrict__ out) {
    __shared__ float red[256];
    const int bnd[6] = {0, 16384, 24576, 32768, 32896, 41088};
    const int p   = blockIdx.x;          // 0..4
    const int tid = threadIdx.x;
    float acc = 0.0f;
    for (int i = bnd[p] + tid; i < bnd[p + 1]; i += 256) acc += w[i];
    red[tid] = acc;
    __syncthreads();
    for (int s = 128; s > 0; s >>= 1) {
        if (tid < s) red[tid] += red[tid + s];
        __syncthreads();
    }
    if (tid == 0) out[p] = red[0];
}

// ---------------------------------------------------------------------------
// Row-sum of a 128-wide f32 segment for 16 consecutive rows, one wave.
// Lane l owns row (l&15); lane-half (l>>4) covers 64 of the 128 columns via
// float4 loads, VALU-accumulated into A.x.  A is the documented 16x4 f32
// A-matrix layout (lanes 0-15 supply K=0,1; lanes 16-31 supply K=2,3), and
// B = all-ones (layout-independent), so after one V_WMMA_F32_16X16X4_F32:
//   D[m][n] = full row-sum of row m, replicated across all N columns.
// D layout: d[r], lanes 0-15 -> row r, lanes 16-31 -> row r+8.
// ---------------------------------------------------------------------------
__device__ __forceinline__ v8f tp_rowsum16(const float* __restrict__ base,
                                           int stride, int lane) {
    const int row  = lane & 15;
    const int half = lane >> 4;
    const float4* p = reinterpret_cast<const float4*>(base + row * stride + half * 64);
    float acc = 0.0f;
#pragma unroll
    for (int j = 0; j < 16; ++j) {
        float4 v = p[j];
        acc += (v.x + v.y) + (v.z + v.w);
    }
    v2f a;    a.x = acc;  a.y = 0.0f;
    v2f ones; ones.x = 1.0f; ones.y = 1.0f;
    v8f c = {};
    c = __builtin_amdgcn_wmma_f32_16x16x4_f32(false, a, false, ones,
                                              (short)0, c, false, false);
    return c;
}

// ---------------------------------------------------------------------------
// Kernel 2: one wave per 16 batch rows.  WMMA row-sums, then coalesced
// streaming of x2 -> y with per-path scalar weights.
// ---------------------------------------------------------------------------
__global__ __launch_bounds__(256) void tp_main(const float* __restrict__ x1,
                                               const float* __restrict__ x2,
                                               const float* __restrict__ wsum,
                                               float* __restrict__ y) {
    const int gwave = (int)((blockIdx.x * blockDim.x + threadIdx.x) >> 5);
    const int lane  = (int)(threadIdx.x & 31);
    if (gwave >= TP_BATCH / 16) return;          // wave-uniform: EXEC all-1s below
    const size_t rowBase = (size_t)gwave * 16;

    const float* x1b = x1 + rowBase * TP_S1;
    const float* x2b = x2 + rowBase * TP_S2;

    // Four independent 128-wide reductions per row (s0,s1,s2 from x1; t0 from x2).
    v8f c0 = tp_rowsum16(x1b + 0,   TP_S1, lane);
    v8f c1 = tp_rowsum16(x1b + 128, TP_S1, lane);
    v8f c2 = tp_rowsum16(x1b + 256, TP_S1, lane);
    v8f c3 = tp_rowsum16(x2b + 0,   TP_S2, lane);

    const float w0 = wsum[0], w1 = wsum[1], w2 = wsum[2], w3 = wsum[3], w4 = wsum[4];

#pragma unroll
    for (int m = 0; m < 16; ++m) {
        const int src = (m < 8) ? 0 : 16;        // lane holding row m's sums
        const int idx = m & 7;                   // D VGPR holding row m
        const float s0 = __shfl(c0[idx], src, 32);
        const float s1 = __shfl(c1[idx], src, 32);
        const float s2 = __shfl(c2[idx], src, 32);
        const float t0 = __shfl(c3[idx], src, 32);

        const float* x2r = x2b + (size_t)m * TP_S2;
        float*       yr  = y + (rowBase + (size_t)m) * TP_SO;

        // y[0:128] = w0*s0 * x2[0:128]  (float4 loads; y only 4B-aligned -> b32 stores)
        const float k0 = w0 * s0;
        float4 v = reinterpret_cast<const float4*>(x2r)[lane];
        yr[lane * 4 + 0] = k0 * v.x;
        yr[lane * 4 + 1] = k0 * v.y;
        yr[lane * 4 + 2] = k0 * v.z;
        yr[lane * 4 + 3] = k0 * v.w;

        // y[128:192] = (w1*s0 + w4*s1) * x2[128:192]
        const float k1 = w1 * s0 + w4 * s1;
        float2 u = reinterpret_cast<const float2*>(x2r + 128)[lane];
        yr[128 + lane * 2 + 0] = k1 * u.x;
        yr[128 + lane * 2 + 1] = k1 * u.y;

        // y[192:256] = w2*s1 * x2[192:256]
        const float k2 = w2 * s1;
        float2 q = reinterpret_cast<const float2*>(x2r + 192)[lane];
        yr[192 + lane * 2 + 0] = k2 * q.x;
        yr[192 + lane * 2 + 1] = k2 * q.y;

        // y[256] = w3 * s2 * t0   ('uuu' path)
        if (lane == 0) yr[256] = w3 * s2 * t0;
    }
}

// ---------------------------------------------------------------------------
extern "C" void kernel_launch(void* const* d_in, const int* in_sizes, int n_in,
                              void* d_out, int out_size, void* d_ws, size_t ws_size,
                              hipStream_t stream) {
    const float* x1     = (const float*)d_in[0];
    const float* x2     = (const float*)d_in[1];
    const float* weight = (const float*)d_in[2];
    float* yv   = (float*)d_out;
    float* wsum = (float*)d_ws;      // 5 floats of scratch

    tp_weight_sums<<<5, 256, 0, stream>>>(weight, wsum);

    const int waves  = TP_BATCH / 16;      // 12500 (exact)
    const int blocks = (waves + 7) / 8;    // 256 threads = 8 wave32s per block
    tp_main<<<blocks, 256, 0, stream>>>(x1, x2, wsum, yv);
}